// KGDDP_85358180041108
// MI455X (gfx1250) — compile-verified
//
#include <hip/hip_runtime.h>

typedef float v2f __attribute__((ext_vector_type(2)));
typedef float v8f __attribute__((ext_vector_type(8)));

#define DD 128

// ---------------------------------------------------------------- utilities
__global__ void fill_zero(float* __restrict__ p, long n) {
  long i = (long)blockIdx.x * blockDim.x + threadIdx.x;
  long stride = (long)gridDim.x * blockDim.x;
  for (; i < n; i += stride) p[i] = 0.0f;
}

// Wsum = sum of 4 [128,128] self-weights; btot = sum of all 8 conv1 biases
__global__ void wsum_kernel(const float* __restrict__ w1, const float* __restrict__ w2,
                            const float* __restrict__ w3, const float* __restrict__ w4,
                            const float* __restrict__ b1, const float* __restrict__ b2,
                            const float* __restrict__ b3, const float* __restrict__ b4,
                            const float* __restrict__ b5, const float* __restrict__ b6,
                            const float* __restrict__ b7, const float* __restrict__ b8,
                            float* __restrict__ Wsum, float* __restrict__ btot) {
  int i = blockIdx.x * blockDim.x + threadIdx.x;
  if (i < 128 * 128) Wsum[i] = w1[i] + w2[i] + w3[i] + w4[i];
  if (i < 128)
    btot[i] = b1[i] + b2[i] + b3[i] + b4[i] + b5[i] + b6[i] + b7[i] + b8[i];
}

// ------------------------------------------------- fp32 WMMA GEMM, N = 128
// C[M,128] = A[M,K] @ W[128,K]^T (+ bias[n]) ; accumulate: C += ...
// grid.x = ceil(M/16), block = 32: one wave owns a 16x128 strip (8 WMMA
// tiles). The A fragment is loaded once per k-step and reused by 8
// independent WMMAs (no D->A/B hazards, 8x A-traffic reduction).
// Out-of-range A rows are clamped, not zeroed: C row m depends only on A
// row m, and rows >= M are never stored.
__global__ void gemm_wmma128(const float* __restrict__ A, const float* __restrict__ W,
                             const float* __restrict__ bias, float* __restrict__ C,
                             int M, int K, int accumulate) {
  const int lane = threadIdx.x;
  const int half = lane >> 4;       // 0: K-offset 0, 1: K-offset 2
  const int l15  = lane & 15;
  const int mbase = blockIdx.x * 16;
  int arow = mbase + l15;
  if (arow >= M) arow = M - 1;      // clamp: stay in-bounds
  const float* Ap = A + (long)arow * K + half * 2;
  const float* Wp = W + (long)l15 * K + half * 2;    // + n*16*K per N tile
  v8f acc0 = {0.f,0.f,0.f,0.f,0.f,0.f,0.f,0.f};
  v8f acc1 = acc0, acc2 = acc0, acc3 = acc0;
  v8f acc4 = acc0, acc5 = acc0, acc6 = acc0, acc7 = acc0;
  for (int k = 0; k < K; k += 4) {
    v2f a = *(const v2f*)(Ap + k);
    v2f b0 = *(const v2f*)(Wp + k);
    v2f b1 = *(const v2f*)(Wp + 16L * K + k);
    v2f b2 = *(const v2f*)(Wp + 32L * K + k);
    v2f b3 = *(const v2f*)(Wp + 48L * K + k);
    v2f b4 = *(const v2f*)(Wp + 64L * K + k);
    v2f b5 = *(const v2f*)(Wp + 80L * K + k);
    v2f b6 = *(const v2f*)(Wp + 96L * K + k);
    v2f b7 = *(const v2f*)(Wp + 112L * K + k);
    acc0 = __builtin_amdgcn_wmma_f32_16x16x4_f32(false, a, false, b0, (short)0, acc0, false, false);
    acc1 = __builtin_amdgcn_wmma_f32_16x16x4_f32(false, a, false, b1, (short)0, acc1, false, false);
    acc2 = __builtin_amdgcn_wmma_f32_16x16x4_f32(false, a, false, b2, (short)0, acc2, false, false);
    acc3 = __builtin_amdgcn_wmma_f32_16x16x4_f32(false, a, false, b3, (short)0, acc3, false, false);
    acc4 = __builtin_amdgcn_wmma_f32_16x16x4_f32(false, a, false, b4, (short)0, acc4, false, false);
    acc5 = __builtin_amdgcn_wmma_f32_16x16x4_f32(false, a, false, b5, (short)0, acc5, false, false);
    acc6 = __builtin_amdgcn_wmma_f32_16x16x4_f32(false, a, false, b6, (short)0, acc6, false, false);
    acc7 = __builtin_amdgcn_wmma_f32_16x16x4_f32(false, a, false, b7, (short)0, acc7, false, false);
  }
  v8f accs[8] = {acc0, acc1, acc2, acc3, acc4, acc5, acc6, acc7};
#pragma unroll
  for (int n = 0; n < 8; ++n) {
    const int col = n * 16 + l15;
    const float bn = bias ? bias[col] : 0.0f;
#pragma unroll
    for (int v = 0; v < 8; ++v) {
      int row = mbase + v + half * 8;  // C/D: vgpr v, lanes>=16 hold rows +8
      if (row < M) {
        float* cp = C + (long)row * DD + col;
        float val = accs[n][v] + bn;
        if (accumulate) val += *cp;
        *cp = val;
      }
    }
  }
}

// ------------------------------------------------- segment mean (SAGE neigh)
// one block of 128 threads per edge: acc[dst] += h_src[src]; cnt[dst] += 1
__global__ void seg_scatter(const float* __restrict__ hsrc, const int* __restrict__ src,
                            const int* __restrict__ dst, float* __restrict__ acc,
                            float* __restrict__ cnt) {
  int e = blockIdx.x;
  int t = threadIdx.x;
  int s = src[e], d = dst[e];
  atomicAdd(&acc[(long)d * DD + t], hsrc[(long)s * DD + t]);
  if (t == 0) atomicAdd(&cnt[d], 1.0f);
}

__global__ void seg_div(float* __restrict__ acc, const float* __restrict__ cnt, int n) {
  int i = blockIdx.x * blockDim.x + threadIdx.x;
  if (i < n) acc[i] = acc[i] / fmaxf(cnt[i >> 7], 1.0f);
}

// ------------------------------------------------- GraphConv norm='both'
__global__ void deg_kernel(const int* __restrict__ src, const int* __restrict__ dst,
                           float* __restrict__ dego, float* __restrict__ degi, int E) {
  int i = blockIdx.x * blockDim.x + threadIdx.x;
  if (i < E) {
    atomicAdd(&dego[src[i]], 1.0f);
    atomicAdd(&degi[dst[i]], 1.0f);
  }
}

__global__ void gc_scatter(const float* __restrict__ h, const int* __restrict__ src,
                           const int* __restrict__ dst, const float* __restrict__ dego,
                           float* __restrict__ agg) {
  int e = blockIdx.x;
  int t = threadIdx.x;
  int s = src[e], d = dst[e];
  float sc = rsqrtf(fmaxf(dego[s], 1.0f));
  atomicAdd(&agg[(long)d * DD + t], h[(long)s * DD + t] * sc);
}

__global__ void gc_norm(float* __restrict__ agg, const float* __restrict__ degi, int n) {
  int i = blockIdx.x * blockDim.x + threadIdx.x;
  if (i < n) agg[i] *= rsqrtf(fmaxf(degi[i >> 7], 1.0f));
}

// ------------------------------------------------- fused MLP scorer
// score(m) = relu(concat(hpro[ipro[m]], hpid[ipid[m]]) @ Wd1^T + bd1) . Wd2
// (bd2 cancels in the h1-h2 / t1-t2 differences). One wave per 16 batch rows;
// sweeps 16 hidden N-tiles, 64 WMMA k-steps each, folds Wd2 dot via shfl_xor.
__global__ void mlp_score(const float* __restrict__ hpro, const float* __restrict__ hpid,
                          const int* __restrict__ ipro, const int* __restrict__ ipid,
                          const float* __restrict__ Wd1, const float* __restrict__ bd1,
                          const float* __restrict__ Wd2, float* __restrict__ out,
                          int subtract) {
  const int lane = threadIdx.x;
  const int half = lane >> 4;
  const int l15  = lane & 15;
  const int mbase = blockIdx.x * 16;
  const int m = mbase + l15;                       // B is a multiple of 16
  const float* rowp = hpro + (long)ipro[m] * DD + half * 2;
  const float* rowq = hpid + (long)ipid[m] * DD + half * 2;
  float rs[8] = {0.f, 0.f, 0.f, 0.f, 0.f, 0.f, 0.f, 0.f};
  for (int nt = 0; nt < 16; ++nt) {
    const int ncol = nt * 16 + l15;
    const float* wrow = Wd1 + (long)ncol * 256 + half * 2;
    v8f acc = {0.f, 0.f, 0.f, 0.f, 0.f, 0.f, 0.f, 0.f};
    for (int k = 0; k < 256; k += 4) {
      const float* ap = (k + half * 2 < 128) ? (rowp + k) : (rowq + k - 128);
      v2f a = *(const v2f*)ap;
      v2f b = *(const v2f*)(wrow + k);
      acc = __builtin_amdgcn_wmma_f32_16x16x4_f32(false, a, false, b, (short)0, acc,
                                                  false, false);
    }
    const float bias = bd1[ncol];
    const float w2   = Wd2[ncol];
#pragma unroll
    for (int v = 0; v < 8; ++v) {
      float hv = fmaxf(acc[v] + bias, 0.0f);
      float p = hv * w2;
      p += __shfl_xor(p, 1);
      p += __shfl_xor(p, 2);
      p += __shfl_xor(p, 4);
      p += __shfl_xor(p, 8);   // sum over the 16 columns held by this half-wave
      rs[v] += p;
    }
  }
  if (l15 < 8) {
    float val = rs[0];
#pragma unroll
    for (int v = 1; v < 8; ++v)
      if (l15 == v) val = rs[v];
    int row = mbase + l15 + half * 8;
    if (subtract) out[row] -= val;
    else          out[row] = val;
  }
}

// ---------------------------------------------------------------- driver
extern "C" void kernel_launch(void* const* d_in, const int* in_sizes, int n_in,
                              void* d_out, int out_size, void* d_ws, size_t ws_size,
                              hipStream_t stream) {
  (void)in_sizes; (void)n_in; (void)out_size; (void)ws_size;
  // inputs (setup_inputs order); pid_fea/W_in_pid/b_in_pid/bd2 are dead code
  const float* drug_fea  = (const float*)d_in[1];
  const float* pro_emb   = (const float*)d_in[2];
  const float* path_emb  = (const float*)d_in[3];
  const float* go_emb    = (const float*)d_in[4];
  const float* W_in_drug = (const float*)d_in[7];
  const float* b_in_drug = (const float*)d_in[8];
  const float* Wn_ppi  = (const float*)d_in[9];
  const float* bn_ppi  = (const float*)d_in[10];
  const float* Ws_ppi  = (const float*)d_in[11];
  const float* bs_ppi  = (const float*)d_in[12];
  const float* Wn_path = (const float*)d_in[13];
  const float* bn_path = (const float*)d_in[14];
  const float* Ws_path = (const float*)d_in[15];
  const float* bs_path = (const float*)d_in[16];
  const float* Wn_go   = (const float*)d_in[17];
  const float* bn_go   = (const float*)d_in[18];
  const float* Ws_go   = (const float*)d_in[19];
  const float* bs_go   = (const float*)d_in[20];
  const float* Wn_dpi  = (const float*)d_in[21];
  const float* bn_dpi  = (const float*)d_in[22];
  const float* Ws_dpi  = (const float*)d_in[23];
  const float* bs_dpi  = (const float*)d_in[24];
  const float* W_gc = (const float*)d_in[25];
  const float* b_gc = (const float*)d_in[26];
  const float* Wd1  = (const float*)d_in[27];
  const float* bd1  = (const float*)d_in[28];
  const float* Wd2  = (const float*)d_in[29];
  const int* ppi_src  = (const int*)d_in[31];
  const int* ppi_dst  = (const int*)d_in[32];
  const int* path_src = (const int*)d_in[33];
  const int* path_dst = (const int*)d_in[34];
  const int* go_src   = (const int*)d_in[35];
  const int* go_dst   = (const int*)d_in[36];
  const int* dpi_src  = (const int*)d_in[37];
  const int* dpi_dst  = (const int*)d_in[38];
  const int* pp_src   = (const int*)d_in[39];
  const int* pp_dst   = (const int*)d_in[40];
  const int* h_head_idx = (const int*)d_in[41];
  const int* h_tail1_id = (const int*)d_in[42];
  const int* h_tail2_id = (const int*)d_in[43];
  const int* t_tail_idx = (const int*)d_in[44];
  const int* t_head1_id = (const int*)d_in[45];
  const int* t_head2_id = (const int*)d_in[46];

  // output layout: h1-h2 | t1-t2 | h_pro | h_pid | dr_in
  float* out_h = (float*)d_out;
  float* out_t = out_h + 50000;
  float* h_pro = out_t + 50000;
  float* h_pid = h_pro + 20000L * DD;
  float* dr_in = h_pid + 1000L * DD;

  // workspace (~10.4 MB)
  float* ws   = (float*)d_ws;
  float* Wsum = ws;                     // 128*128
  float* btot = ws + 16384;             // 128
  float* acc  = ws + 16512;             // 20000*128 (segment-sum / hn buffer)
  float* cnt  = acc + 20000L * DD;      // 20000 (contiguous after acc)
  float* agg  = acc;                    // reuse after relations: 1000*128
  float* dego = acc + 128000;           // 20000
  float* degi = acc + 148000;           // 1000

  // 1) combined self-weight + bias (4 Ws + all 8 conv1 biases)
  wsum_kernel<<<64, 256, 0, stream>>>(Ws_ppi, Ws_path, Ws_go, Ws_dpi,
                                      bn_ppi, bs_ppi, bn_path, bs_path,
                                      bn_go, bs_go, bn_dpi, bs_dpi, Wsum, btot);

  // 2) dr_in = drug_fea @ W_in_drug^T + b   [8000,1024]x[128,1024]^T
  gemm_wmma128<<<500, 32, 0, stream>>>(drug_fea, W_in_drug, b_in_drug, dr_in,
                                       8000, 1024, 0);

  // 3) h_pro = pro_emb @ (sum Ws)^T + sum bias
  gemm_wmma128<<<1250, 32, 0, stream>>>(pro_emb, Wsum, btot, h_pro,
                                        20000, 128, 0);

  // 4) four SAGE relations: seg-mean then accumulate hn @ Wn^T into h_pro
  const float* hs[4]  = {pro_emb, path_emb, go_emb, dr_in};
  const int*   srA[4] = {ppi_src, path_src, go_src, dpi_src};
  const int*   dsA[4] = {ppi_dst, path_dst, go_dst, dpi_dst};
  const float* WnA[4] = {Wn_ppi, Wn_path, Wn_go, Wn_dpi};
  const int    EA[4]  = {500000, 300000, 400000, 250000};
  for (int r = 0; r < 4; ++r) {
    fill_zero<<<1024, 256, 0, stream>>>(acc, 20000L * DD + 20000);
    seg_scatter<<<EA[r], DD, 0, stream>>>(hs[r], srA[r], dsA[r], acc, cnt);
    seg_div<<<(20000 * DD + 255) / 256, 256, 0, stream>>>(acc, cnt, 20000 * DD);
    gemm_wmma128<<<1250, 32, 0, stream>>>(acc, WnA[r], nullptr, h_pro,
                                          20000, 128, 1);
  }

  // 5) GraphConv(norm='both') protein -> pid
  fill_zero<<<256, 256, 0, stream>>>(agg, 149000L);
  deg_kernel<<<(300000 + 255) / 256, 256, 0, stream>>>(pp_src, pp_dst, dego, degi, 300000);
  gc_scatter<<<300000, DD, 0, stream>>>(h_pro, pp_src, pp_dst, dego, agg);
  gc_norm<<<(1000 * DD + 255) / 256, 256, 0, stream>>>(agg, degi, 1000 * DD);
  gemm_wmma128<<<63, 32, 0, stream>>>(agg, W_gc, b_gc, h_pid, 1000, 128, 0);

  // 6) MLP scoring: out = score(a) - score(b)  (bd2 cancels)
  mlp_score<<<3125, 32, 0, stream>>>(h_pro, h_pid, h_head_idx, h_tail1_id,
                                     Wd1, bd1, Wd2, out_h, 0);
  mlp_score<<<3125, 32, 0, stream>>>(h_pro, h_pid, h_head_idx, h_tail2_id,
                                     Wd1, bd1, Wd2, out_h, 1);
  mlp_score<<<3125, 32, 0, stream>>>(h_pro, h_pid, t_head1_id, t_tail_idx,
                                     Wd1, bd1, Wd2, out_t, 0);
  mlp_score<<<3125, 32, 0, stream>>>(h_pro, h_pid, t_head2_id, t_tail_idx,
                                     Wd1, bd1, Wd2, out_t, 1);
}